// ForwardTransformLayer_59983513256495
// MI455X (gfx1250) — compile-verified
//
#include <hip/hip_runtime.h>
#include <stdint.h>

// CDNA5 / gfx1250: wave32, WMMA 16x16 shapes, async global->LDS staging.
typedef __attribute__((ext_vector_type(2))) float v2f;
typedef __attribute__((ext_vector_type(8))) float v8f;

#define M_COLS 8192     // input row length
#define P_LEN  4096     // output row length (part_length)
#define TS     140      // LDS row stride (floats): 16B-aligned rows, conflict-free A reads
#define RT     4        // row tiles processed per block (double-buffered)

// One DWT level: details + approximation = two 8-tap circular stride-2 filters,
// formulated as a banded GEMM on the f32 WMMA path, with GLOBAL_LOAD_ASYNC_TO_LDS_B128
// double-buffered staging (ASYNCcnt pipeline).
__global__ __launch_bounds__(128) void dwt_wmma_kernel(const float* __restrict__ x,
                                                       const float* __restrict__ wavelet,
                                                       float* __restrict__ out) {
    __shared__ float tile[2][16 * TS];

    const int tid  = threadIdx.x;     // 0..127 (4 waves of 32)
    const int lane = tid & 31;
    const int wave = tid >> 5;        // 0..3 : which 16-output k-subtile
    const int kk   = lane & 15;       // N index within 16x16 tile
    const int half = lane >> 4;       // 0: lanes 0-15, 1: lanes 16-31

    const int k0   = blockIdx.x * 64; // first output column of this block
    const int rgrp = blockIdx.y * RT; // first row-tile index of this block

    // ---------------- filter taps ----------------
    // details taps: g_d[t] = w[6 - 2*(t>>1) + (t&1)]   (t = offset-from-(2k-6))
    // approx  taps: same index map into s[i] = (i&1 ? -w[7-i] : w[7-i])
    float gd[8], ga[8];
    {
        float w[8];
#pragma unroll
        for (int i = 0; i < 8; ++i) w[i] = wavelet[i];
#pragma unroll
        for (int t = 0; t < 8; ++t) {
            const int idx = 6 - 2 * (t >> 1) + (t & 1);     // compile-time constant
            gd[t] = w[idx];
            ga[t] = (idx & 1) ? -w[7 - idx] : w[7 - idx];   // s[idx]
        }
    }

    // ---------------- banded weight matrix W (identical for every tile) ----------------
    // W[n, kk] = g[n - 2*kk] when 0 <= n-2*kk <= 7, else 0;  n = 0..39 in ten K=4 chunks.
    // B fragment layout (4x16 f32): comp j, lane L -> W[4c + 2*half + j, L&15].
    v2f Bd[10], Ba[10];
#pragma unroll
    for (int c = 0; c < 10; ++c) {
        v2f bd, ba;
#pragma unroll
        for (int j = 0; j < 2; ++j) {
            const int u = 4 * c + 2 * half + j - 2 * kk;    // tap index
            float vd = 0.0f, va = 0.0f;
#pragma unroll
            for (int t = 0; t < 8; ++t) {                   // select chain, stays in VGPRs
                const bool m = (u == t);
                vd = m ? gd[t] : vd;
                va = m ? ga[t] : va;
            }
            bd[j] = vd; ba[j] = va;
        }
        Bd[c] = bd; Ba[c] = ba;
    }

    // ---------------- async staging: global -> LDS, 16B-aligned float4 groups ----------
    // Staged window starts at cbase = 2*k0 - 8 (== 0 mod 4, so every float4 is 16B aligned
    // and, since 8192 % 4 == 0, never straddles the circular wrap). Outputs need columns
    // cbase+2 .. cbase+137 inside the 140-wide tile.
    const int cbase = 2 * k0 - 8 + M_COLS;                  // keep non-negative before mask
    const unsigned lds0 = (unsigned)(uintptr_t)(&tile[0][0]);   // wave-relative LDS offset
    const unsigned lds1 = (unsigned)(uintptr_t)(&tile[1][0]);

    auto prefetch = [&](int rtile, unsigned ldsbase) {
        // 16 rows x 35 float4 = 560 groups; clamp so every wave issues exactly 5 ops
        // (duplicates of the last group are benign) -> ASYNCcnt bookkeeping is uniform.
#pragma unroll
        for (int it = 0; it < 5; ++it) {
            int i = tid + it * 128;
            i = (i > 559) ? 559 : i;
            const int row = i / 35;
            const int c4  = i - row * 35;
            const unsigned gcol = (unsigned)(cbase + 4 * c4) & (M_COLS - 1);
            const unsigned goff = ((((unsigned)(rtile * 16 + row)) << 13) + gcol) << 2;
            const unsigned lds  = ldsbase + (unsigned)(row * TS + 4 * c4) * 4u;
            asm volatile("global_load_async_to_lds_b128 %0, %1, %2"
                         :: "v"(lds), "v"(goff), "s"(x)
                         : "memory");
        }
    };

    prefetch(rgrp, lds0);                                   // pipeline prologue

#pragma unroll
    for (int rt = 0; rt < RT; ++rt) {
        // Prefetch next tile into the other buffer, then wait for the current tile.
        if (rt + 1 < RT) {
            prefetch(rgrp + rt + 1, (rt & 1) ? lds0 : lds1);
            asm volatile("s_wait_asynccnt 5" ::: "memory"); // current tile done, next in flight
        } else {
            asm volatile("s_wait_asynccnt 0" ::: "memory");
        }
        __syncthreads();

        // ---------------- banded GEMM: ten V_WMMA_F32_16X16X4_F32 per output tile ----------
        // A fragment (16x4 f32): comp j, lane L ->
        //   X[r0 + (L&15), local 2 + 32*wave + 4c + 2*half + j]
        const float* lrow = &tile[rt & 1][(lane & 15) * TS + 2 + 32 * wave + 2 * half];
        v8f accd = {};
        v8f acca = {};
#pragma unroll
        for (int c = 0; c < 10; ++c) {
            v2f a;
            a[0] = lrow[4 * c + 0];
            a[1] = lrow[4 * c + 1];
            accd = __builtin_amdgcn_wmma_f32_16x16x4_f32(false, a, false, Bd[c],
                                                         (short)0, accd, false, false);
            acca = __builtin_amdgcn_wmma_f32_16x16x4_f32(false, a, false, Ba[c],
                                                         (short)0, acca, false, false);
        }

        // ---------------- store C/D fragments ----------------
        // VGPR i, lanes 0-15 -> row r0+i, col kout; lanes 16-31 -> row r0+8+i.
        const int r0   = (rgrp + rt) * 16;
        const int kout = k0 + 16 * wave + kk;
#pragma unroll
        for (int i = 0; i < 8; ++i) {
            const size_t ro = (size_t)(r0 + i + 8 * half) * P_LEN + kout;
            out[ro] = accd[i];                              // details
            out[(size_t)P_LEN * P_LEN + ro] = acca[i];      // approximation
        }

        __syncthreads();   // all waves done reading tile[rt&1] before it is refilled
    }
}

extern "C" void kernel_launch(void* const* d_in, const int* in_sizes, int n_in,
                              void* d_out, int out_size, void* d_ws, size_t ws_size,
                              hipStream_t stream) {
    const float* x       = (const float*)d_in[0];   // (4096, 8192) f32
    const float* wavelet = (const float*)d_in[1];   // (8,) f32
    float* out           = (float*)d_out;           // details (4096x4096) then approx

    dim3 grid(P_LEN / 64, 4096 / (16 * RT));        // (64, 64)
    dwt_wmma_kernel<<<grid, 128, 0, stream>>>(x, wavelet, out);
}